// Decoder_21706764714369
// MI455X (gfx1250) — compile-verified
//
#include <hip/hip_runtime.h>
#include <math.h>

typedef float v2f __attribute__((ext_vector_type(2)));
typedef float v8f __attribute__((ext_vector_type(8)));

#define N_NODES 50000
#define BB 32
#define TT 512
#define SD 64
#define OT 128
#define GATES 512          // 4*OT
#define BT (BB * TT)       // 16384

__device__ __forceinline__ v8f wmma_f32(v2f a, v2f b, v8f c) {
    // V_WMMA_F32_16X16X4_F32 : D = A(16x4) * B(4x16) + C(16x16), all fp32
    return __builtin_amdgcn_wmma_f32_16x16x4_f32(
        /*neg_a=*/false, a, /*neg_b=*/false, b,
        /*c_mod=*/(short)0, c, /*reuse_a=*/false, /*reuse_b=*/false);
}

// ---------------------------------------------------------------- utilities
__global__ void zero_f32(float* __restrict__ p, long n) {
    long i = (long)blockIdx.x * blockDim.x + threadIdx.x;
    if (i < n) p[i] = 0.0f;
}

__global__ void degree_kernel(const int* __restrict__ src, const int* __restrict__ dst,
                              float* __restrict__ deg_out, float* __restrict__ deg_in, int E) {
    int e = blockIdx.x * blockDim.x + threadIdx.x;
    if (e < E) {
        atomicAdd(&deg_out[src[e]], 1.0f);
        atomicAdd(&deg_in[dst[e]], 1.0f);
    }
}

__global__ void norm_finish(float* __restrict__ no, float* __restrict__ ni, int n) {
    int i = blockIdx.x * blockDim.x + threadIdx.x;
    if (i < n) {
        no[i] = rsqrtf(fmaxf(no[i], 1.0f));
        ni[i] = rsqrtf(fmaxf(ni[i], 1.0f));
    }
}

// layer-1 message + aggregate: agg1[dst] += concat(ht,c_hs)[src] * norm_out[src]
__global__ void scatter1(const int* __restrict__ src, const int* __restrict__ dst,
                         const float* __restrict__ ht, const float* __restrict__ chs,
                         const float* __restrict__ norm_out, float* __restrict__ agg1, int E) {
    long tid = (long)blockIdx.x * blockDim.x + threadIdx.x;   // E*32 threads
    int e = (int)(tid >> 5);
    if (e >= E) return;
    int q = ((int)tid & 31) << 2;                             // k base, 0..124, step 4
    int s = src[e], d = dst[e];
    float no = norm_out[s];
    const float* sp = (q < SD) ? (ht + (long)s * SD + q) : (chs + (long)s * SD + (q - SD));
    float4 v = *(const float4*)sp;
    float* op = agg1 + (long)d * 128 + q;
    atomicAdd(op + 0, v.x * no);
    atomicAdd(op + 1, v.y * no);
    atomicAdd(op + 2, v.z * no);
    atomicAdd(op + 3, v.w * no);
}

// layer-2 message + aggregate: agg2[dst] += x1[src] * norm_out[src]   (64 wide)
__global__ void scatter2(const int* __restrict__ src, const int* __restrict__ dst,
                         const float* __restrict__ x1, const float* __restrict__ norm_out,
                         float* __restrict__ agg2, int E) {
    long tid = (long)blockIdx.x * blockDim.x + threadIdx.x;   // E*16 threads
    int e = (int)(tid >> 4);
    if (e >= E) return;
    int q = ((int)tid & 15) << 2;
    int s = src[e], d = dst[e];
    float no = norm_out[s];
    float4 v = *(const float4*)(x1 + (long)s * 64 + q);
    float* op = agg2 + (long)d * 64 + q;
    atomicAdd(op + 0, v.x * no);
    atomicAdd(op + 1, v.y * no);
    atomicAdd(op + 2, v.z * no);
    atomicAdd(op + 3, v.w * no);
}

// W[512][128] row-major -> WT[128][512]
__global__ void transpose_k(const float* __restrict__ W, float* __restrict__ WT) {
    int idx = blockIdx.x * blockDim.x + threadIdx.x;  // 512*128
    if (idx >= GATES * OT) return;
    int r = idx >> 7, c = idx & 127;
    WT[c * GATES + r] = W[idx];
}

__global__ void bias_sum_k(const float* __restrict__ a, const float* __restrict__ b,
                           float* __restrict__ o) {
    int i = blockIdx.x * blockDim.x + threadIdx.x;
    if (i < GATES) o[i] = a[i] + b[i];
}

// Xcat[row][0:64]=hs[row], [64:128]=c_ht[row]   row = b*T + t
__global__ void concat_k(const float* __restrict__ hs, const float* __restrict__ cht,
                         float* __restrict__ X) {
    int idx = blockIdx.x * blockDim.x + threadIdx.x;  // BT*128
    if (idx >= BT * 128) return;
    int row = idx >> 7, k = idx & 127;
    X[idx] = (k < SD) ? hs[row * SD + k] : cht[row * SD + (k - SD)];
}

// ------------------------------------------------- generic WMMA fp32 GEMM
// C[M,N] = act( (A * rowscale) @ B + bias ), tiles exact multiples of 16, K%4==0
__global__ void gemm_wmma_kernel(const float* __restrict__ A, const float* __restrict__ rowscale,
                                 const float* __restrict__ Bm, const float* __restrict__ bias,
                                 float* __restrict__ C, int M, int N, int K, int applyGelu) {
    const int wave = threadIdx.x >> 5;
    const int lane = threadIdx.x & 31;
    const int hl = lane >> 4;        // 0 or 1
    const int l16 = lane & 15;
    const int ct = blockIdx.y * (blockDim.x >> 5) + wave;
    if ((ct << 4) >= N) return;      // wave-uniform; EXEC stays all-ones inside
    const int row0 = blockIdx.x << 4;
    const int col = (ct << 4) + l16;
    const int arow = row0 + l16;
    const float rs = rowscale ? rowscale[arow] : 1.0f;
    const float* Arow = A + (size_t)arow * K;
    v8f acc = {};
    for (int k = 0; k < K; k += 4) {
        const int ka = k + (hl << 1);
        v2f a, b;
        a.x = Arow[ka] * rs;
        a.y = Arow[ka + 1] * rs;
        b.x = Bm[(size_t)ka * N + col];
        b.y = Bm[(size_t)(ka + 1) * N + col];
        acc = wmma_f32(a, b, acc);
    }
    const float bv = bias ? bias[col] : 0.0f;
#pragma unroll
    for (int r = 0; r < 8; ++r) {
        float v = acc[r] + bv;
        if (applyGelu) v = 0.5f * v * (1.0f + erff(v * 0.70710678118654752f));
        C[(size_t)(row0 + r + (hl << 3)) * N + col] = v;
    }
}

// ------------------------------------------------- persistent LSTM recurrence
// One workgroup (32 waves on one WGP). Per step: gates[32,512] = h[32,128] @ WhhT
// via WMMA f32 16x16x4. Wave w owns column tile w; its B tile (128x16 fp32 =
// 64 floats/lane) is register-resident for the whole sequence, so the critical
// path per step is: ds_load h -> 64 WMMAs -> LDS gate shuffle -> activations.
// gates_pre for the current step is prefetched into registers before the GEMM
// so the global loads overlap the matrix work. Two barriers per step.
__global__ __launch_bounds__(1024) void lstm_kernel(const float* __restrict__ gpre,  // [B*T][512]
                                                    const float* __restrict__ WhhT,  // [128][512]
                                                    float* __restrict__ out) {       // [B][T][128]
    __shared__ float h_lds[BB][OT];        // 16 KB
    __shared__ float g_lds[BB][GATES];     // 64 KB
    const int tid = threadIdx.x;
    const int wave = tid >> 5, lane = tid & 31;
    const int hl = lane >> 4, l16 = lane & 15;
    const int col0 = wave << 4;            // 32 waves cover N=512
    const int col = col0 + l16;

    // B tile -> registers, once (2 floats per k-step, 32 k-steps)
    float bx[32], by[32];
#pragma unroll
    for (int kk = 0; kk < 32; ++kk) {
        const int ka = (kk << 2) + (hl << 1);
        bx[kk] = WhhT[ka * GATES + col];
        by[kk] = WhhT[(ka + 1) * GATES + col];
    }

    for (int i = tid; i < BB * OT; i += 1024) ((float*)h_lds)[i] = 0.0f;
    float creg[4] = {0.f, 0.f, 0.f, 0.f};  // each thread owns 4 (b,j) cells
    __syncthreads();

    for (int t = 0; t < TT; ++t) {
        // prefetch this step's input-projection gates (overlaps with GEMM below)
        float gp[4][4];
#pragma unroll
        for (int u = 0; u < 4; ++u) {
            const int p = (tid << 2) + u;          // 0..4095
            const int b = p >> 7, j = p & 127;
            const float* g = gpre + ((long)b * TT + t) * GATES + j;
            gp[u][0] = g[0];
            gp[u][1] = g[OT];
            gp[u][2] = g[2 * OT];
            gp[u][3] = g[3 * OT];
        }

        v8f c0 = {}, c1 = {};              // row tiles m=0 (b 0..15), m=1 (b 16..31)
#pragma unroll
        for (int kk = 0; kk < 32; ++kk) {
            const int ka = (kk << 2) + (hl << 1);
            v2f b, a0, a1;
            b.x = bx[kk];                b.y = by[kk];
            a0.x = h_lds[l16][ka];       a0.y = h_lds[l16][ka + 1];
            a1.x = h_lds[16 + l16][ka];  a1.y = h_lds[16 + l16][ka + 1];
            c0 = wmma_f32(a0, b, c0);
            c1 = wmma_f32(a1, b, c1);
        }
#pragma unroll
        for (int r = 0; r < 8; ++r) {
            g_lds[r + (hl << 3)][col]      = c0[r];
            g_lds[16 + r + (hl << 3)][col] = c1[r];
        }
        __syncthreads();   // all h_lds reads (GEMM) and g_lds writes complete

#pragma unroll
        for (int u = 0; u < 4; ++u) {
            const int p = (tid << 2) + u;
            const int b = p >> 7, j = p & 127;
            float gi = g_lds[b][j]          + gp[u][0];
            float gf = g_lds[b][OT + j]     + gp[u][1];
            float gg = g_lds[b][2 * OT + j] + gp[u][2];
            float go = g_lds[b][3 * OT + j] + gp[u][3];
            float si = 1.0f / (1.0f + expf(-gi));
            float sf = 1.0f / (1.0f + expf(-gf));
            float so = 1.0f / (1.0f + expf(-go));
            float cc = sf * creg[u] + si * tanhf(gg);
            creg[u] = cc;
            float hh = so * tanhf(cc);
            h_lds[b][j] = hh;              // safe: everyone passed the barrier above
            out[((long)b * TT + t) * OT + j] = hh;
        }
        __syncthreads();   // new h visible before next step's GEMM
    }
}

// ---------------------------------------------------------------- launcher
extern "C" void kernel_launch(void* const* d_in, const int* in_sizes, int n_in,
                              void* d_out, int out_size, void* d_ws, size_t ws_size,
                              hipStream_t stream) {
    const float* hs  = (const float*)d_in[0];
    const float* ht  = (const float*)d_in[1];
    const float* chs = (const float*)d_in[2];
    const float* cht = (const float*)d_in[3];
    const int*   src = (const int*)d_in[4];
    const int*   dst = (const int*)d_in[5];
    const float* W1  = (const float*)d_in[6];
    const float* b1  = (const float*)d_in[7];
    const float* W2  = (const float*)d_in[8];
    const float* b2  = (const float*)d_in[9];
    const float* Wih = (const float*)d_in[10];
    const float* Whh = (const float*)d_in[11];
    const float* bih = (const float*)d_in[12];
    const float* bhh = (const float*)d_in[13];
    const int E = in_sizes[4];

    // workspace layout (floats); zeroed region first
    float* ws       = (float*)d_ws;
    float* norm_out = ws;                                    // N
    float* norm_in  = norm_out + N_NODES;                    // N
    float* agg1     = norm_in + N_NODES;                     // N*128
    float* agg2     = agg1 + (size_t)N_NODES * 128;          // N*64
    float* x1       = agg2 + (size_t)N_NODES * 64;           // N*64
    float* Xcat     = x1 + (size_t)N_NODES * 64;             // BT*128
    float* gpre     = Xcat + (size_t)BT * 128;               // BT*512
    float* WihT     = gpre + (size_t)BT * GATES;             // 128*512
    float* WhhT     = WihT + OT * GATES;                     // 128*512
    float* bsum     = WhhT + OT * GATES;                     // 512

    const int thr = 256;
    const long zn = (long)N_NODES * (2 + 128 + 64);
    zero_f32<<<(int)((zn + thr - 1) / thr), thr, 0, stream>>>(norm_out, zn);

    degree_kernel<<<(E + thr - 1) / thr, thr, 0, stream>>>(src, dst, norm_out, norm_in, E);
    norm_finish<<<(N_NODES + thr - 1) / thr, thr, 0, stream>>>(norm_out, norm_in, N_NODES);

    const long t1 = (long)E * 32;
    scatter1<<<(int)((t1 + thr - 1) / thr), thr, 0, stream>>>(src, dst, ht, chs, norm_out, agg1, E);

    dim3 gs(N_NODES / 16, 1);   // 3125 row tiles x (4 waves = 4 col tiles)
    gemm_wmma_kernel<<<gs, 128, 0, stream>>>(agg1, norm_in, W1, b1, x1, N_NODES, 64, 128, 1);

    const long t2 = (long)E * 16;
    scatter2<<<(int)((t2 + thr - 1) / thr), thr, 0, stream>>>(src, dst, x1, norm_out, agg2, E);

    float* out_sp = (float*)d_out + (size_t)BT * OT;         // spatial output after [B,T,OT]
    gemm_wmma_kernel<<<gs, 128, 0, stream>>>(agg2, norm_in, W2, b2, out_sp, N_NODES, 64, 64, 1);

    transpose_k<<<(GATES * OT + thr - 1) / thr, thr, 0, stream>>>(Wih, WihT);
    transpose_k<<<(GATES * OT + thr - 1) / thr, thr, 0, stream>>>(Whh, WhhT);
    bias_sum_k<<<2, 256, 0, stream>>>(bih, bhh, bsum);
    concat_k<<<(BT * 128 + thr - 1) / thr, thr, 0, stream>>>(hs, cht, Xcat);

    dim3 gp(BT / 16, 8);        // 1024 row tiles x (8 blocks * 4 waves = 32 col tiles)
    gemm_wmma_kernel<<<gp, 128, 0, stream>>>(Xcat, nullptr, WihT, bsum, gpre, BT, GATES, 128, 0);

    lstm_kernel<<<1, 1024, 0, stream>>>(gpre, WhhT, (float*)d_out);
}